// HGNN_20246475833495
// MI455X (gfx1250) — compile-verified
//
#include <hip/hip_runtime.h>

#define N_NODES 50000
#define N_EDGES 5000
#define NNZ     800000
#define DIM     256

typedef float v2f __attribute__((ext_vector_type(2)));
typedef float v8f __attribute__((ext_vector_type(8)));

// ---------------------------------------------------------------------------
// Degree accumulation: one pass over the incidence list, both degree arrays.
// ---------------------------------------------------------------------------
__global__ void degree_kernel(const int* __restrict__ nidx,
                              const int* __restrict__ eidx,
                              float* __restrict__ deg_n,
                              float* __restrict__ deg_e) {
    int i = blockIdx.x * blockDim.x + threadIdx.x;
    if (i < NNZ) {
        atomicAdd(&deg_n[nidx[i]], 1.0f);
        atomicAdd(&deg_e[eidx[i]], 1.0f);
    }
}

__global__ void invert_kernel(float* __restrict__ d, int n) {
    int i = blockIdx.x * blockDim.x + threadIdx.x;
    if (i < n) {
        float v = d[i];
        d[i] = (v > 0.0f) ? (1.0f / v) : 0.0f;
    }
}

// ---------------------------------------------------------------------------
// GEMM: out[M,256] = A[M,256] @ W[256,256] using V_WMMA_F32_16X16X4_F32.
// Block = 128 threads = 4 waves, covering 16 rows x 256 cols. Each wave owns
// a 16x64 strip held in FOUR independent accumulators:
//   - one A fragment feeds 4 wmma per k-step (4x A reuse)
//   - 4 independent D->C chains keep the matrix pipe busy
// grid = (M/16) 1-D; no divergence anywhere (EXEC all-1s, required by WMMA).
// A-fragment (16x4):  lane<16 -> K = {k, k+1} with k=k0; lane>=16 -> k=k0+2
// B-fragment (4x16):  same K split; N = lane & 15
// C/D: VGPR r -> row (r + 8*(lane>>4)), col (lane & 15)
// ---------------------------------------------------------------------------
__global__ void gemm_wmma_f32(const float* __restrict__ A,
                              const float* __restrict__ W,
                              float* __restrict__ out) {
    const int wave = threadIdx.x >> 5;
    const int lane = threadIdx.x & 31;
    const int half = lane >> 4;   // 0 or 1
    const int l16  = lane & 15;

    const int row0 = blockIdx.x * 16;
    const int col0 = wave * 64;

    const float* __restrict__ Arow = A + (size_t)(row0 + l16) * DIM;
    const float* __restrict__ Bbase = W + col0 + l16;

    v8f c0 = {0.f,0.f,0.f,0.f,0.f,0.f,0.f,0.f};
    v8f c1 = c0, c2 = c0, c3 = c0;

    #pragma unroll 2
    for (int k0 = 0; k0 < DIM; k0 += 4) {
        const int k = k0 + 2 * half;
        v2f a;
        a.x = Arow[k + 0];
        a.y = Arow[k + 1];
        const float* __restrict__ Bk0 = Bbase + (size_t)(k + 0) * DIM;
        const float* __restrict__ Bk1 = Bbase + (size_t)(k + 1) * DIM;
        v2f b0, b1, b2, b3;
        b0.x = Bk0[ 0]; b0.y = Bk1[ 0];
        b1.x = Bk0[16]; b1.y = Bk1[16];
        b2.x = Bk0[32]; b2.y = Bk1[32];
        b3.x = Bk0[48]; b3.y = Bk1[48];
        c0 = __builtin_amdgcn_wmma_f32_16x16x4_f32(false, a, false, b0, (short)0, c0, false, false);
        c1 = __builtin_amdgcn_wmma_f32_16x16x4_f32(false, a, false, b1, (short)0, c1, false, false);
        c2 = __builtin_amdgcn_wmma_f32_16x16x4_f32(false, a, false, b2, (short)0, c2, false, false);
        c3 = __builtin_amdgcn_wmma_f32_16x16x4_f32(false, a, false, b3, (short)0, c3, false, false);
    }

    float* __restrict__ orow = out + (size_t)(row0 + 8 * half) * DIM + col0 + l16;
    #pragma unroll
    for (int r = 0; r < 8; ++r) {
        orow[(size_t)r * DIM +  0] = c0[r];
        orow[(size_t)r * DIM + 16] = c1[r];
        orow[(size_t)r * DIM + 32] = c2[r];
        orow[(size_t)r * DIM + 48] = c3[r];
    }
}

// ---------------------------------------------------------------------------
// Gather row src[gidx[i]] (optionally scaled by rscale[gidx[i]]) and
// atomically scatter-add into dst[sidx[i]]. 256 threads handle 4 incidence
// rows (64 float4 lanes per row).
// ---------------------------------------------------------------------------
__global__ void gather_scatter_add(const float* __restrict__ src,
                                   const float* __restrict__ rscale,
                                   const int* __restrict__ gidx,
                                   const int* __restrict__ sidx,
                                   float* __restrict__ dst) {
    int t  = blockIdx.x * blockDim.x + threadIdx.x;
    int i  = t >> 6;               // incidence entry
    int c  = (t & 63) << 2;        // float4 column
    if (i >= NNZ) return;
    int g = gidx[i];
    int s = sidx[i];
    float sc = rscale ? rscale[g] : 1.0f;
    const float4 v = *reinterpret_cast<const float4*>(src + (size_t)g * DIM + c);
    float* d = dst + (size_t)s * DIM + c;
    atomicAdd(d + 0, v.x * sc);
    atomicAdd(d + 1, v.y * sc);
    atomicAdd(d + 2, v.z * sc);
    atomicAdd(d + 3, v.w * sc);
}

// ---------------------------------------------------------------------------
// h = relu(acc * Dinv[node] + bias)   (elementwise, float4)
// ---------------------------------------------------------------------------
__global__ void finalize_relu(const float* __restrict__ acc,
                              const float* __restrict__ dinv,
                              const float* __restrict__ bias,
                              float* __restrict__ h) {
    int t = blockIdx.x * blockDim.x + threadIdx.x;
    if (t >= N_NODES * (DIM / 4)) return;
    int node = t >> 6;
    int c    = (t & 63) << 2;
    float s  = dinv[node];
    float4 v = *reinterpret_cast<const float4*>(acc + (size_t)node * DIM + c);
    float4 b = *reinterpret_cast<const float4*>(bias + c);
    float4 o;
    o.x = fmaxf(fmaf(v.x, s, b.x), 0.0f);
    o.y = fmaxf(fmaf(v.y, s, b.y), 0.0f);
    o.z = fmaxf(fmaf(v.z, s, b.z), 0.0f);
    o.w = fmaxf(fmaf(v.w, s, b.w), 0.0f);
    *reinterpret_cast<float4*>(h + (size_t)node * DIM + c) = o;
}

// ---------------------------------------------------------------------------
// Fused layer-2 finalize + head: out[n] = sum_c relu(acc*D + b2)[c]*Wc[c] + bc
// One wave (32 lanes) per node, shuffle reduction.
// ---------------------------------------------------------------------------
__global__ void finalize_project(const float* __restrict__ acc,
                                 const float* __restrict__ dinv,
                                 const float* __restrict__ b2,
                                 const float* __restrict__ Wc,
                                 const float* __restrict__ bc,
                                 float* __restrict__ out) {
    int wave = threadIdx.x >> 5;
    int lane = threadIdx.x & 31;
    int node = blockIdx.x * 8 + wave;
    if (node >= N_NODES) return;
    float s = dinv[node];
    const float* row = acc + (size_t)node * DIM;
    float sum = 0.0f;
    #pragma unroll
    for (int c = lane; c < DIM; c += 32) {
        float v = fmaxf(fmaf(row[c], s, b2[c]), 0.0f);
        sum = fmaf(v, Wc[c], sum);
    }
    #pragma unroll
    for (int o = 16; o > 0; o >>= 1)
        sum += __shfl_xor(sum, o, 32);
    if (lane == 0) out[node] = sum + bc[0];
}

// ---------------------------------------------------------------------------
extern "C" void kernel_launch(void* const* d_in, const int* in_sizes, int n_in,
                              void* d_out, int out_size, void* d_ws, size_t ws_size,
                              hipStream_t stream) {
    const float* x    = (const float*)d_in[0];
    const int*   nidx = (const int*)  d_in[1];
    const int*   eidx = (const int*)  d_in[2];
    const float* W1   = (const float*)d_in[3];
    const float* b1   = (const float*)d_in[4];
    const float* W2   = (const float*)d_in[5];
    const float* b2   = (const float*)d_in[6];
    const float* Wc   = (const float*)d_in[7];
    const float* bc   = (const float*)d_in[8];
    float* out = (float*)d_out;

    // Workspace layout (floats)
    float* P    = (float*)d_ws;                       // N*256
    float* Q    = P    + (size_t)N_NODES * DIM;       // N*256
    float* Mb   = Q    + (size_t)N_NODES * DIM;       // E*256
    float* Dinv = Mb   + (size_t)N_EDGES * DIM;       // N
    float* Binv = Dinv + N_NODES;                     // E

    const size_t szN  = (size_t)N_NODES * DIM * sizeof(float);
    const size_t szE  = (size_t)N_EDGES * DIM * sizeof(float);

    dim3 gemm_grid(N_NODES / 16);
    dim3 gemm_blk(128);
    const int sc_blocks = (NNZ * 64 + 255) / 256;         // gather/scatter
    const int ew_blocks = (N_NODES * 64 + 255) / 256;     // elementwise N*256/4

    // --- degrees (shared by both layers) ---
    hipMemsetAsync(Dinv, 0, (size_t)(N_NODES + N_EDGES) * sizeof(float), stream);
    degree_kernel<<<(NNZ + 255) / 256, 256, 0, stream>>>(nidx, eidx, Dinv, Binv);
    invert_kernel<<<(N_NODES + N_EDGES + 255) / 256, 256, 0, stream>>>(Dinv, N_NODES + N_EDGES);

    // --- layer 1 ---
    gemm_wmma_f32<<<gemm_grid, gemm_blk, 0, stream>>>(x, W1, P);          // P = x @ W1
    hipMemsetAsync(Mb, 0, szE, stream);
    gather_scatter_add<<<sc_blocks, 256, 0, stream>>>(P, nullptr, nidx, eidx, Mb);
    hipMemsetAsync(Q, 0, szN, stream);
    gather_scatter_add<<<sc_blocks, 256, 0, stream>>>(Mb, Binv, eidx, nidx, Q);
    finalize_relu<<<ew_blocks, 256, 0, stream>>>(Q, Dinv, b1, P);         // P = h1

    // --- layer 2 ---
    gemm_wmma_f32<<<gemm_grid, gemm_blk, 0, stream>>>(P, W2, Q);          // Q = h1 @ W2
    hipMemsetAsync(Mb, 0, szE, stream);
    gather_scatter_add<<<sc_blocks, 256, 0, stream>>>(Q, nullptr, nidx, eidx, Mb);
    hipMemsetAsync(P, 0, szN, stream);
    gather_scatter_add<<<sc_blocks, 256, 0, stream>>>(Mb, Binv, eidx, nidx, P);

    // --- fused finalize + head ---
    finalize_project<<<(N_NODES + 7) / 8, 256, 0, stream>>>(P, Dinv, b2, Wc, bc, out);
}